// GatedGCNLayer_46986942218355
// MI455X (gfx1250) — compile-verified
//
#include <hip/hip_runtime.h>
#include <math.h>

// GatedGCN layer for MI455X (gfx1250, wave32, WMMA + TDM).
// d_out = [h_out (N*D) | e_out (E*D)], fp32.

typedef __attribute__((ext_vector_type(2))) float    v2f;
typedef __attribute__((ext_vector_type(4))) float    f4;
typedef __attribute__((ext_vector_type(8))) float    v8f;
typedef __attribute__((ext_vector_type(4))) unsigned u32x4;
typedef __attribute__((ext_vector_type(8))) unsigned u32x8;

#define DIM 128
#define PITCH 132          // LDS row pitch in floats: 132 % 64 == 4 -> conflict-free b64 frag loads
#define BN_EPS 1e-5f

// ---------------------------------------------------------------------------
// WMMA helper: D = A(16x4) * B(4x16) + C, fp32 matrix core.
// ---------------------------------------------------------------------------
__device__ __forceinline__ v8f wmma_f32_k4(v2f a, v2f b, v8f c) {
    // 8 args: (neg_a, A, neg_b, B, c_mod, C, reuse_a, reuse_b)
    return __builtin_amdgcn_wmma_f32_16x16x4_f32(false, a, false, b, (short)0, c,
                                                 false, false);
}

// Fragment load from an LDS-resident row-major tile (pitch = PITCH floats).
// A layout (16x4 f32): lanes 0-15 hold M=lane, {K=4kk, 4kk+1}; lanes 16-31 hold
// M=lane-16, {K=4kk+2, 4kk+3}.  B (4x16) uses the same pattern with N=lane&15.
__device__ __forceinline__ v2f frag_ld(const float* base, int lane, int kk) {
    const int r  = lane & 15;
    const int k0 = (kk << 2) + ((lane >> 4) << 1);
    return *(const v2f*)(base + r * PITCH + k0);
}

// ---------------------------------------------------------------------------
// TDM: issue one tensor_load_to_lds moving a 16x128 f32 tile (row-major,
// stride 128) from global memory into LDS at byte offset `ldsByteOff`,
// inserting 4 pad DWORDs after every 128 DWORDs (-> PITCH=132 rows in LDS).
// 2D tensor -> 2 descriptor groups (VADDR2/3 = NULL form). Tracked by
// TENSORcnt. Caller must be a uniform (whole-wave) context.
// ---------------------------------------------------------------------------
__device__ __forceinline__ void tdm_load_tile16x128(const float* gsrc,
                                                    unsigned ldsByteOff) {
    const unsigned long long ga = (unsigned long long)(const void*)gsrc;
    u32x4 g0;
    g0[0] = 1u;                                        // count=1, user descriptor
    g0[1] = ldsByteOff;                                // lds_addr (bytes)
    g0[2] = (unsigned)(ga & 0xFFFFFFFFu);              // global_addr[31:0]
    g0[3] = (unsigned)((ga >> 32) & 0x1FFFFFFu)        // global_addr[56:32]
          | (2u << 30);                                // type = 2 (image/D#)
    u32x8 g1;
    g1[0] = (2u << 16)        // data_size = 2 -> 4-byte elements
          | (1u << 20)        // pad_enable
          | (6u << 22)        // pad_interval code 6 -> every 128 DWORDs
          | (3u << 25);       // pad_amount  code 3 -> 4 DWORDs
    g1[1] = (128u << 16);     // tensor_dim0 = 128  (bits 79:48)
    g1[2] = (16u  << 16);     // tensor_dim1 = 16   (bits 111:80)
    g1[3] = (128u << 16);     // tile_dim0 = 128    (bits 127:112)
    g1[4] = 16u;              // tile_dim1 = 16     (bits 143:128)
    g1[5] = 128u;             // tensor_dim0_stride = 128 (bits 207:160)
    g1[6] = 0u;
    g1[7] = 0u;
    asm volatile("tensor_load_to_lds %0, %1" :: "s"(g0), "s"(g1) : "memory");
}

// ---------------------------------------------------------------------------
// Kernel 1: Ah/Bh/Dh/Eh = h @ W{a,b,d,e}.T + b{a,b,d,e}
// Block = 256 threads = 8 wave32s; wave w owns output columns [16w, 16w+16).
// ---------------------------------------------------------------------------
__global__ __launch_bounds__(256) void gg_node_gemm(
    const float* __restrict__ h,
    const float* __restrict__ Wa, const float* __restrict__ ba,
    const float* __restrict__ Wb, const float* __restrict__ bb,
    const float* __restrict__ Wd, const float* __restrict__ bd,
    const float* __restrict__ We, const float* __restrict__ be,
    float* __restrict__ Ah, float* __restrict__ Bh,
    float* __restrict__ Dh, float* __restrict__ Eh, int nRows)
{
    extern __shared__ float smem[];
    float* Wl = smem;                 // DIM * PITCH floats
    float* Tl = smem + DIM * PITCH;   // 16  * PITCH floats

    const int tid  = threadIdx.x;
    const int lane = tid & 31;
    const int wv   = tid >> 5;                 // 0..7: column tile
    const int col  = wv * 16 + (lane & 15);

    const float* Wlist[4] = {Wa, Wb, Wd, We};
    const float* blist[4] = {ba, bb, bd, be};
    float*       olist[4] = {Ah, Bh, Dh, Eh};

    const int nTiles = nRows >> 4;

#pragma unroll
    for (int m = 0; m < 4; ++m) {
        __syncthreads();                       // retire readers of previous W
        {   // stage W[m] (row-major 128x128) into padded LDS, float4 copies
            const float* Wg = Wlist[m];
            const int row = tid >> 1;
            const int cb  = (tid & 1) << 6;
#pragma unroll
            for (int j = 0; j < 16; ++j) {
                f4 v = *(const f4*)(Wg + row * DIM + cb + j * 4);
                *(f4*)(Wl + row * PITCH + cb + j * 4) = v;
            }
        }
        __syncthreads();

        const float bias = blist[m][col];
        float* outp = olist[m];
        const float* Bbase = Wl + (wv * 16) * PITCH;

        for (int tile = blockIdx.x; tile < nTiles; tile += gridDim.x) {
            {   // stage 16x128 h tile
                const int row = tid >> 4;
                const int cb  = (tid & 15) << 3;
                const float* gp = h + (tile * 16 + row) * DIM + cb;
                f4 v0 = *(const f4*)(gp);
                f4 v1 = *(const f4*)(gp + 4);
                *(f4*)(Tl + row * PITCH + cb)     = v0;
                *(f4*)(Tl + row * PITCH + cb + 4) = v1;
            }
            __syncthreads();

            v8f acc = {};
#pragma unroll
            for (int kk = 0; kk < 32; ++kk) {
                v2f a = frag_ld(Tl, lane, kk);
                v2f b = frag_ld(Bbase, lane, kk);
                acc = wmma_f32_k4(a, b, acc);
            }

            // C/D layout: vgpr r -> row r (lanes 0-15) / r+8 (lanes 16-31)
            const int rbase = tile * 16 + ((lane >> 4) << 3);
#pragma unroll
            for (int r = 0; r < 8; ++r)
                outp[(rbase + r) * DIM + col] = acc[r] + bias;

            __syncthreads();                  // before next tile overwrites Tl
        }
    }
}

// ---------------------------------------------------------------------------
// Kernel 2: fused edge stage, TDM double-buffered.
// Per 16-edge tile: Ce = e_tile @ Wc.T (WMMA, Wc in LDS), then
//   e_new = Ce + bc + Dh[src] + Eh[dst];  sigma = sigmoid(e_new)
//   e_out = relu(bn_e(e_new));  num[dst] += sigma*Bh[src];  den[dst] += sigma
// e tiles are staged LDS-ward by the Tensor Data Mover (TENSORcnt), with the
// next tile's DMA in flight while the current tile computes.
// ---------------------------------------------------------------------------
__global__ __launch_bounds__(256) void gg_edge_fused(
    const float* __restrict__ e,
    const int* __restrict__ src, const int* __restrict__ dst,
    const float* __restrict__ Wc, const float* __restrict__ bc,
    const float* __restrict__ Bh, const float* __restrict__ Dh,
    const float* __restrict__ Eh,
    const float* __restrict__ gamma_e, const float* __restrict__ beta_e,
    const float* __restrict__ mean_e,  const float* __restrict__ var_e,
    float* __restrict__ num, float* __restrict__ den,
    float* __restrict__ e_out, int nEdges)
{
    extern __shared__ float smem[];
    float* Wl = smem;                                  // DIM * PITCH
    // two TDM destination buffers, 16 * PITCH floats each
    const unsigned tileOff0 = (unsigned)(DIM * PITCH) * sizeof(float);
    const unsigned tileBytes = (unsigned)(16 * PITCH) * sizeof(float);

    const int tid  = threadIdx.x;
    const int lane = tid & 31;
    const int wv   = tid >> 5;
    const int col  = wv * 16 + (lane & 15);

    // Per-column constants live in registers for the whole kernel.
    const float bias = bc[col];
    const float sc   = gamma_e[col] * rsqrtf(var_e[col] + BN_EPS);
    const float sh   = beta_e[col] - mean_e[col] * sc;

    {   // stage Wc into padded LDS
        const int row = tid >> 1;
        const int cb  = (tid & 1) << 6;
#pragma unroll
        for (int j = 0; j < 16; ++j) {
            f4 v = *(const f4*)(Wc + row * DIM + cb + j * 4);
            *(f4*)(Wl + row * PITCH + cb + j * 4) = v;
        }
    }

    const int nTiles = nEdges >> 4;
    const int stride = gridDim.x;
    const float* Bbase = Wl + (wv * 16) * PITCH;

    int cur = 0;
    // Prologue: kick off the DMA for this block's first tile.
    if (wv == 0 && blockIdx.x < nTiles)
        tdm_load_tile16x128(e + (size_t)blockIdx.x * 16 * DIM, tileOff0);

    for (int tile = blockIdx.x; tile < nTiles; tile += stride, cur ^= 1) {
        const int nxt = tile + stride;
        if (wv == 0) {
            if (nxt < nTiles) {
                // Prefetch next tile into the other buffer, then ensure the
                // current tile (older TDM op) has landed: TENSORcnt <= 1.
                tdm_load_tile16x128(e + (size_t)nxt * 16 * DIM,
                                    tileOff0 + (cur ^ 1) * tileBytes);
                __builtin_amdgcn_s_wait_tensorcnt(1);
            } else {
                __builtin_amdgcn_s_wait_tensorcnt(0);
            }
        }
        __syncthreads();   // buf[cur] (and Wc on iter 0) visible to all waves

        const float* Tl = smem + DIM * PITCH + cur * 16 * PITCH;
        v8f acc = {};
#pragma unroll
        for (int kk = 0; kk < 32; ++kk) {
            v2f a = frag_ld(Tl, lane, kk);
            v2f b = frag_ld(Bbase, lane, kk);
            acc = wmma_f32_k4(a, b, acc);
        }

        const int rsel = (lane >> 4) << 3;
#pragma unroll
        for (int r = 0; r < 8; ++r) {
            const int er   = rsel + r;
            const int eidx = tile * 16 + er;
            const int s = src[eidx];      // 16 distinct values/tile, L0-broadcast
            const int d = dst[eidx];
            const float en  = acc[r] + bias + Dh[s * DIM + col] + Eh[d * DIM + col];
            const float sig = 1.0f / (1.0f + __expf(-en));
            const float bh  = Bh[s * DIM + col];
            e_out[eidx * DIM + col] = fmaxf(en * sc + sh, 0.0f);
            __hip_atomic_fetch_add(&num[d * DIM + col], sig * bh,
                                   __ATOMIC_RELAXED, __HIP_MEMORY_SCOPE_AGENT);
            __hip_atomic_fetch_add(&den[d * DIM + col], sig,
                                   __ATOMIC_RELAXED, __HIP_MEMORY_SCOPE_AGENT);
        }
        __syncthreads();   // all readers of buf[cur] done before TDM reuses it
    }
}

// ---------------------------------------------------------------------------
// Kernel 3: h_out = relu(bn_h(Ah + num/(den+1e-6))), float4-vectorized.
// ---------------------------------------------------------------------------
__global__ __launch_bounds__(256) void gg_node_final(
    const float* __restrict__ Ah, const float* __restrict__ num,
    const float* __restrict__ den,
    const float* __restrict__ gamma_h, const float* __restrict__ beta_h,
    const float* __restrict__ mean_h,  const float* __restrict__ var_h,
    float* __restrict__ h_out, int total4)
{
    const int i = blockIdx.x * blockDim.x + threadIdx.x;
    if (i >= total4) return;
    const int c = (i & 31) << 2;              // column base (DIM/4 == 32)
    f4 a  = *(const f4*)(Ah  + i * 4);
    f4 nm = *(const f4*)(num + i * 4);
    f4 dn = *(const f4*)(den + i * 4);
    f4 g  = *(const f4*)(gamma_h + c);
    f4 b  = *(const f4*)(beta_h + c);
    f4 mu = *(const f4*)(mean_h + c);
    f4 vv = *(const f4*)(var_h + c);
    f4 o;
#pragma unroll
    for (int j = 0; j < 4; ++j) {
        const float hn = a[j] + nm[j] / (dn[j] + 1e-6f);
        const float s  = g[j] * rsqrtf(vv[j] + BN_EPS);
        o[j] = fmaxf((hn - mu[j]) * s + b[j], 0.0f);
    }
    *(f4*)(h_out + i * 4) = o;
}

// ---------------------------------------------------------------------------
extern "C" void kernel_launch(void* const* d_in, const int* in_sizes, int n_in,
                              void* d_out, int out_size, void* d_ws, size_t ws_size,
                              hipStream_t stream)
{
    const float* h   = (const float*)d_in[0];
    const float* e   = (const float*)d_in[1];
    const int*   src = (const int*)d_in[2];
    const int*   dst = (const int*)d_in[3];
    const float* Wa  = (const float*)d_in[4];
    const float* ba  = (const float*)d_in[5];
    const float* Wb  = (const float*)d_in[6];
    const float* bb  = (const float*)d_in[7];
    const float* Wc  = (const float*)d_in[8];
    const float* bc  = (const float*)d_in[9];
    const float* Wd  = (const float*)d_in[10];
    const float* bd  = (const float*)d_in[11];
    const float* We  = (const float*)d_in[12];
    const float* be  = (const float*)d_in[13];
    const float* gamma_h = (const float*)d_in[14];
    const float* beta_h  = (const float*)d_in[15];
    const float* mean_h  = (const float*)d_in[16];
    const float* var_h   = (const float*)d_in[17];
    const float* gamma_e = (const float*)d_in[18];
    const float* beta_e  = (const float*)d_in[19];
    const float* mean_e  = (const float*)d_in[20];
    const float* var_e   = (const float*)d_in[21];

    const int N = in_sizes[0] / DIM;   // 20000
    const int E = in_sizes[2];         // 640000

    // Workspace: Ah | Bh | Dh | Eh | num | den  (6 * N * DIM floats ~ 61 MB)
    float* Ah  = (float*)d_ws;
    float* Bh  = Ah  + (size_t)N * DIM;
    float* Dh  = Bh  + (size_t)N * DIM;
    float* Eh  = Dh  + (size_t)N * DIM;
    float* num = Eh  + (size_t)N * DIM;
    float* den = num + (size_t)N * DIM;

    float* h_out = (float*)d_out;
    float* e_out = h_out + (size_t)N * DIM;

    // num/den must be zero every call (atomic accumulation).
    hipMemsetAsync(num, 0, (size_t)2 * N * DIM * sizeof(float), stream);

    const size_t smem1 = (size_t)(DIM * PITCH + 16 * PITCH) * sizeof(float);
    const size_t smem2 = (size_t)(DIM * PITCH + 2 * 16 * PITCH) * sizeof(float);

    gg_node_gemm<<<dim3(416), dim3(256), smem1, stream>>>(
        h, Wa, ba, Wb, bb, Wd, bd, We, be, Ah, Bh, Dh, Eh, N);

    gg_edge_fused<<<dim3(2048), dim3(256), smem2, stream>>>(
        e, src, dst, Wc, bc, Bh, Dh, Eh,
        gamma_e, beta_e, mean_e, var_e, num, den, e_out, E);

    const int total4 = N * DIM / 4;
    gg_node_final<<<dim3((total4 + 255) / 256), dim3(256), 0, stream>>>(
        Ah, num, den, gamma_h, beta_h, mean_h, var_h, h_out, total4);
}